// GCN_16355235463694
// MI455X (gfx1250) — compile-verified
//
#include <hip/hip_runtime.h>

typedef __attribute__((ext_vector_type(2))) float     v2f;
typedef __attribute__((ext_vector_type(8))) float     v8f;
typedef __attribute__((ext_vector_type(2))) long long v2i64;

// ---------------------------------------------------------------------------
// Zero the accumulators (out-sum lives directly in d_out, degree in ws).
// ---------------------------------------------------------------------------
__global__ void gmm_zero(float* __restrict__ outSum, float* __restrict__ deg, int n) {
  int i = blockIdx.x * blockDim.x + threadIdx.x;
  if (i < n) { outSum[i] = 0.0f; deg[i] = 0.0f; }
}

// ---------------------------------------------------------------------------
// WMMA precompute: P[n*K + k] = x[n,0]*w[k,0] + x[n,1]*w[k,1]
// One wave per 16-node tile; V_WMMA_F32_16X16X4_F32, features padded 2->4,
// two B tiles cover k=0..K-1 (K<=32).  CDNA5 ISA 7.12.2 layouts:
//   A (16x4 f32): lanes 0-15 -> M=lane, V0=K0, V1=K1; lanes 16-31 -> K2,K3 (zero)
//   B (4x16 f32): lanes 0-15 -> N=lane, V0=K0 row, V1=K1 row; upper lanes zero
//   C/D (16x16):  VGPR v, lane l -> M = v + (l<16?0:8), N = l&15
// ---------------------------------------------------------------------------
__global__ __launch_bounds__(256) void gmm_precompute_wmma(
    const float* __restrict__ x, const float* __restrict__ w,
    float* __restrict__ P, int nNodes, int K) {
  const int lane = threadIdx.x & 31;
  const int wid  = (blockIdx.x * blockDim.x + threadIdx.x) >> 5;
  const int base = wid * 16;
  if (base >= nNodes) return;           // wave-uniform: EXEC all-ones at WMMA

  const int sub = lane & 15;

  v2f a = {0.0f, 0.0f};                 // K=2,3 lanes stay zero (feature padding)
  if (lane < 16) {
    int node = base + sub;
    if (node < nNodes) { a.x = x[2 * node]; a.y = x[2 * node + 1]; }
  }

  v2f b0 = {0.0f, 0.0f};
  v2f b1 = {0.0f, 0.0f};
  if (lane < 16) {
    if (sub < K)      { b0.x = w[2 * sub];  b0.y = w[2 * sub + 1]; }
    int k2 = 16 + sub;
    if (k2 < K)       { b1.x = w[2 * k2];   b1.y = w[2 * k2 + 1]; }
  }

  v8f c = {};
  v8f d0 = __builtin_amdgcn_wmma_f32_16x16x4_f32(false, a, false, b0, (short)0, c, false, false);
  v8f d1 = __builtin_amdgcn_wmma_f32_16x16x4_f32(false, a, false, b1, (short)0, c, false, false);

  // 32-bit addressing: node*K <= nNodes*K ~ 5M, fits int comfortably.
  const int  rowOff = (lane < 16) ? 0 : 8;
  const int  node0  = base + rowOff;
  const bool okLo   = (sub < K);        // loop-invariant, hoisted
  const bool okHi   = (16 + sub < K);
  float* __restrict__ p0 = P + node0 * K + sub;
#pragma unroll
  for (int v = 0; v < 8; ++v) {
    if (node0 + v < nNodes) {
      if (okLo) p0[v * K]      = d0[v];
      if (okHi) p0[v * K + 16] = d1[v];
    }
  }
}

// ---------------------------------------------------------------------------
// Edge scatter, 2 edges/thread: b128 index loads, non-temporal streaming so the
// one-pass edge data (151 MB) does not evict the L2-resident P table (20 MB).
// ---------------------------------------------------------------------------
__global__ __launch_bounds__(256) void gmm_edges_table_x2(
    const v2i64* __restrict__ eRow2, const v2i64* __restrict__ eCol2,
    const v2f* __restrict__ pseudo2, const float* __restrict__ P,
    float* __restrict__ outSum, float* __restrict__ deg,
    int nPairs, int K) {
  const float scale = (float)(K - 1);
  int stride = gridDim.x * blockDim.x;
  for (int i = blockIdx.x * blockDim.x + threadIdx.x; i < nPairs; i += stride) {
    v2i64 r2 = __builtin_nontemporal_load(&eRow2[i]);
    v2i64 c2 = __builtin_nontemporal_load(&eCol2[i]);
    v2f   p2 = __builtin_nontemporal_load(&pseudo2[i]);
#pragma unroll
    for (int j = 0; j < 2; ++j) {
      int r = (int)r2[j];
      int c = (int)c2[j];
      float v    = p2[j] * scale;
      float lo   = floorf(v);
      float frac = v - lo;
      int i0 = min(max((int)lo, 0), K - 1);
      int i1 = min(i0 + 1, K - 1);
      const float* pr = P + c * K;           // regular-temporal: L2 hit
      float m0 = pr[i0];
      float m1 = pr[i1];
      float msg = fmaf(frac, m1 - m0, m0);   // (1-frac)*m0 + frac*m1
      unsafeAtomicAdd(&outSum[r], msg);      // native global_atomic_add_f32
      unsafeAtomicAdd(&deg[r], 1.0f);
    }
  }
}

// Scalar variant (odd edge count).
__global__ __launch_bounds__(256) void gmm_edges_table(
    const long long* __restrict__ eRow, const long long* __restrict__ eCol,
    const float* __restrict__ pseudo, const float* __restrict__ P,
    float* __restrict__ outSum, float* __restrict__ deg,
    int nEdges, int K) {
  const float scale = (float)(K - 1);
  int stride = gridDim.x * blockDim.x;
  for (int i = blockIdx.x * blockDim.x + threadIdx.x; i < nEdges; i += stride) {
    int r = (int)__builtin_nontemporal_load(&eRow[i]);
    int c = (int)__builtin_nontemporal_load(&eCol[i]);
    float v    = __builtin_nontemporal_load(&pseudo[i]) * scale;
    float lo   = floorf(v);
    float frac = v - lo;
    int i0 = min(max((int)lo, 0), K - 1);
    int i1 = min(i0 + 1, K - 1);
    const float* pr = P + c * K;
    float msg = fmaf(frac, pr[i1] - pr[i0], pr[i0]);
    unsafeAtomicAdd(&outSum[r], msg);
    unsafeAtomicAdd(&deg[r], 1.0f);
  }
}

// ---------------------------------------------------------------------------
// Fallback (workspace too small for the table): direct per-edge dot,
// weights staged once per block in LDS.
// ---------------------------------------------------------------------------
__global__ __launch_bounds__(256) void gmm_edges_direct(
    const long long* __restrict__ eRow, const long long* __restrict__ eCol,
    const float* __restrict__ pseudo, const float* __restrict__ x,
    const float* __restrict__ w,
    float* __restrict__ outSum, float* __restrict__ deg,
    int nEdges, int K) {
  extern __shared__ float wsh[];       // 2*K floats
  for (int t = threadIdx.x; t < 2 * K; t += blockDim.x) wsh[t] = w[t];
  __syncthreads();
  const float scale = (float)(K - 1);
  int stride = gridDim.x * blockDim.x;
  for (int i = blockIdx.x * blockDim.x + threadIdx.x; i < nEdges; i += stride) {
    int r = (int)eRow[i];
    int c = (int)eCol[i];
    float v    = pseudo[i] * scale;
    float lo   = floorf(v);
    float frac = v - lo;
    int i0 = min(max((int)lo, 0), K - 1);
    int i1 = min(i0 + 1, K - 1);
    float x0 = x[2 * c], x1 = x[2 * c + 1];
    float m0 = x0 * wsh[2 * i0] + x1 * wsh[2 * i0 + 1];
    float m1 = x0 * wsh[2 * i1] + x1 * wsh[2 * i1 + 1];
    float msg = fmaf(frac, m1 - m0, m0);
    unsafeAtomicAdd(&outSum[r], msg);
    unsafeAtomicAdd(&deg[r], 1.0f);
  }
}

// ---------------------------------------------------------------------------
// Finalize: mean with degree clamped to >= 1.
// ---------------------------------------------------------------------------
__global__ void gmm_finalize(float* __restrict__ outSum, const float* __restrict__ deg, int n) {
  int i = blockIdx.x * blockDim.x + threadIdx.x;
  if (i < n) outSum[i] = outSum[i] / fmaxf(deg[i], 1.0f);
}

// ---------------------------------------------------------------------------
extern "C" void kernel_launch(void* const* d_in, const int* in_sizes, int n_in,
                              void* d_out, int out_size, void* d_ws, size_t ws_size,
                              hipStream_t stream) {
  const float*     x      = (const float*)d_in[0];            // (N, 2)
  const float*     weight = (const float*)d_in[1];            // (K, 2, 1)
  const long long* edges  = (const long long*)d_in[2];        // (2, E) int64
  const float*     pseudo = (const float*)d_in[3];            // (E, 1)

  const int nNodes = in_sizes[0] / 2;
  const int K      = in_sizes[1] / 2;
  const int nEdges = in_sizes[3];

  const long long* eRow = edges;
  const long long* eCol = edges + nEdges;

  float* outSum = (float*)d_out;

  // Workspace layout: [0, N*4) degree; [1 MiB, 1 MiB + N*K*4) projection table.
  const size_t degBytes = (size_t)nNodes * sizeof(float);
  const size_t pOff     = 1u << 20;
  const size_t pBytes   = (size_t)nNodes * (size_t)K * sizeof(float);
  float* deg = (float*)d_ws;
  float* P   = (float*)((char*)d_ws + pOff);
  const bool useTable = (K <= 32) && (ws_size >= pOff + pBytes) && (pOff >= degBytes);

  // 1) Zero accumulators.
  {
    int blk = 256, grd = (nNodes + blk - 1) / blk;
    gmm_zero<<<grd, blk, 0, stream>>>(outSum, deg, nNodes);
  }

  if (useTable) {
    // 2) WMMA precompute of P = X * W^T (one wave per 16-node tile).
    int tiles = (nNodes + 15) / 16;
    int grd   = (tiles + 7) / 8;                 // 8 waves (256 thr) per block
    gmm_precompute_wmma<<<grd, 256, 0, stream>>>(x, weight, P, nNodes, K);

    // 3) Edge scatter with table lookup.
    if ((nEdges & 1) == 0) {
      int nPairs = nEdges >> 1;
      int egrd = (nPairs + 255) / 256;
      if (egrd > 32768) egrd = 32768;
      gmm_edges_table_x2<<<egrd, 256, 0, stream>>>(
          (const v2i64*)eRow, (const v2i64*)eCol, (const v2f*)pseudo,
          P, outSum, deg, nPairs, K);
    } else {
      int egrd = (nEdges + 255) / 256;
      if (egrd > 32768) egrd = 32768;
      gmm_edges_table<<<egrd, 256, 0, stream>>>(eRow, eCol, pseudo, P,
                                                outSum, deg, nEdges, K);
    }
  } else {
    int egrd = (nEdges + 255) / 256;
    if (egrd > 32768) egrd = 32768;
    gmm_edges_direct<<<egrd, 256, 2 * K * sizeof(float), stream>>>(
        eRow, eCol, pseudo, x, weight, outSum, deg, nEdges, K);
  }

  // 4) Mean normalization.
  {
    int blk = 256, grd = (nNodes + blk - 1) / blk;
    gmm_finalize<<<grd, blk, 0, stream>>>(outSum, deg, nNodes);
  }
}